// YOLOELoss_30270929502993
// MI455X (gfx1250) — compile-verified
//
#include <hip/hip_runtime.h>
#include <hip/hip_bf16.h>
#include <math.h>

typedef float v2f __attribute__((ext_vector_type(2)));
typedef float v8f __attribute__((ext_vector_type(8)));

#define NCLS  80
#define NBINS 17
#define MAXA  8448   // A=8400 padded to multiple of 256/16
#define KMAX  13

// ---------------------------------------------------------------- zero ws
__global__ void zero_ws_kernel(unsigned int* ws, int n) {
    int i = blockIdx.x * blockDim.x + threadIdx.x;
    if (i < n) ws[i] = 0u;
}

// ---------------------------------------------------------------- assign
// One block (256 threads = 8 waves) per (b, g).
// d2[a] computed with V_WMMA_F32_16X16X4_F32:
//   A-row(m)  = [ax, ay, ax^2+ay^2, 1]
//   B-col(n)  = [-2cx, -2cy, 1, cx^2+cy^2]   (all 16 columns identical)
//   D[m,n]    = (ax-cx)^2 + (ay-cy)^2
__global__ void assign_kernel(const float* __restrict__ gt_boxes,
                              const float* __restrict__ anchor_points,
                              unsigned int* __restrict__ mask,
                              int B, int G, int A, int K)
{
    __shared__ float d2[MAXA];
    __shared__ float rv[256];
    __shared__ int   ri[256];
    __shared__ int   topk[32];
    __shared__ int   insideArr[32];
    __shared__ int   anyFlag;

    const int bg  = blockIdx.x;
    const int b   = bg / G, g = bg % G;
    const int tid = threadIdx.x;
    const int lane = tid & 31;
    const int wid  = tid >> 5;

    const float* gb = gt_boxes + ((size_t)b * G + g) * 4;
    const float x0 = gb[0], y0 = gb[1], x1 = gb[2], y1 = gb[3];
    const float cx = 0.5f * (x0 + x1), cy = 0.5f * (y0 + y1);
    const float c2 = cx * cx + cy * cy;

    // ---- distances via WMMA (EXEC all-ones: uniform loop, no divergence)
    const int TILES = MAXA / 16;
    {
        v2f Bm;
        if (lane < 16) { Bm.x = -2.0f * cx; Bm.y = -2.0f * cy; } // K=0, K=1
        else           { Bm.x = 1.0f;       Bm.y = c2;         } // K=2, K=3
        for (int t = wid; t < TILES; t += 8) {
            int m0 = t * 16;
            int ai = m0 + (lane & 15);
            int ac = ai < A ? ai : (A - 1);
            float ax = anchor_points[2 * ac + 0];
            float ay = anchor_points[2 * ac + 1];
            v2f Am;
            if (lane < 16) { Am.x = ax;                Am.y = ay;   } // K=0,1
            else           { Am.x = ax * ax + ay * ay; Am.y = 1.0f; } // K=2,3
            v8f C = {};
            C = __builtin_amdgcn_wmma_f32_16x16x4_f32(
                    false, Am, false, Bm, (short)0, C, false, false);
            // C/D layout: VGPR r -> M=r (lanes 0-15) / M=r+8 (lanes 16-31)
            if ((lane & 15) == 0) {
                int base = m0 + (lane >> 4) * 8;
#pragma unroll
                for (int r = 0; r < 8; ++r) d2[base + r] = C[r];
            }
        }
    }
    __syncthreads();
    for (int a = A + tid; a < MAXA; a += 256) d2[a] = 1e30f;  // pad
    __syncthreads();

    // ---- K rounds of argmin (lowest index wins ties, like lax.top_k)
    for (int r = 0; r < K; ++r) {
        float best = 1e30f; int bidx = MAXA;
        for (int a = tid; a < MAXA; a += 256) {
            float v = d2[a];
            if (v < best) { best = v; bidx = a; }
        }
        rv[tid] = best; ri[tid] = bidx;
        __syncthreads();
        for (int s = 128; s > 0; s >>= 1) {
            if (tid < s) {
                float vo = rv[tid + s]; int io = ri[tid + s];
                if (vo < rv[tid] || (vo == rv[tid] && io < ri[tid])) {
                    rv[tid] = vo; ri[tid] = io;
                }
            }
            __syncthreads();
        }
        if (tid == 0) { topk[r] = ri[0]; d2[ri[0]] = 1e30f; }
        __syncthreads();
    }

    // ---- inside test + assign_k fallback + bitmask scatter
    if (tid < K) {
        int a = topk[tid];
        float ax = anchor_points[2 * a + 0];
        float ay = anchor_points[2 * a + 1];
        insideArr[tid] = (ax >= x0) && (ax <= x1) && (ay >= y0) && (ay <= y1);
    }
    __syncthreads();
    if (tid == 0) {
        int any = 0;
        for (int i = 0; i < K; ++i) any |= insideArr[i];
        anyFlag = any;
    }
    __syncthreads();
    if (tid < K) {
        int asg = anyFlag ? insideArr[tid] : (tid < 3);
        if (asg) atomicOr(&mask[(size_t)b * A + topk[tid]], 1u << g);
    }
}

// ---------------------------------------------------------------- loss
// One thread per (b, a); float4 (b128) streaming loads for the 43 MB
// cls tensor (bandwidth-bound at 23.3 TB/s: this kernel is the roofline).
__global__ void loss_kernel(const float* __restrict__ cls,
                            const float* __restrict__ reg,
                            const float* __restrict__ gt_boxes,
                            const int*   __restrict__ gt_labels,
                            const float* __restrict__ anchor_points,
                            const float* __restrict__ stride_t,
                            const unsigned int* __restrict__ mask,
                            float* accum, unsigned int* cnt,
                            int B, int G, int A)
{
    __shared__ float sp[256], sn[256], sb[256];
    __shared__ unsigned int sc[256];
    __shared__ int   slbl[32];
    __shared__ float sgtb[128];

    const int b   = blockIdx.y;
    const int tid = threadIdx.x;
    const int a   = blockIdx.x * 256 + tid;

    if (tid < G)     slbl[tid] = gt_labels[b * G + tid];
    if (tid < 4 * G) sgtb[tid] = gt_boxes[((size_t)b * G) * 4 + tid];
    __syncthreads();

    float pos = 0.f, neg = 0.f, box = 0.f; unsigned int c = 0;

    if (a < A) {
        unsigned int m = mask[(size_t)b * A + a];
        int fg = (m != 0u);

        // target_cls = union of one-hot labels over assigned GTs
        unsigned int lm[3] = {0u, 0u, 0u};
        unsigned int mm = m;
        while (mm) {
            int g = __ffs((int)mm) - 1;
            mm &= mm - 1u;
            int lbl = slbl[g];
            lm[lbl >> 5] |= 1u << (lbl & 31);
        }

        // ---- BCE over 80 classes, b128 loads (20 x float4 per anchor)
        const float4* pc4 = (const float4*)(cls + ((size_t)b * A + a) * NCLS);
        float s = 0.f;
#pragma unroll 5
        for (int v = 0; v < NCLS / 4; ++v) {
            float4 pv = pc4[v];
            int cc = v * 4;
            unsigned bits = (lm[cc >> 5] >> (cc & 31)) & 0xFu;
            float p0 = fminf(fmaxf(pv.x, 1e-7f), 1.0f - 1e-7f);
            float p1 = fminf(fmaxf(pv.y, 1e-7f), 1.0f - 1e-7f);
            float p2 = fminf(fmaxf(pv.z, 1e-7f), 1.0f - 1e-7f);
            float p3 = fminf(fmaxf(pv.w, 1e-7f), 1.0f - 1e-7f);
            s -= logf((bits & 1u) ? p0 : 1.0f - p0);
            s -= logf((bits & 2u) ? p1 : 1.0f - p1);
            s -= logf((bits & 4u) ? p2 : 1.0f - p2);
            s -= logf((bits & 8u) ? p3 : 1.0f - p3);
        }
        if (fg) { pos = s; c = 1; } else { neg = s; }

        // ---- DFL box loss (fg anchors only; 17 x float4 per anchor)
        if (fg) {
            int last = 31 - __clz((int)m);   // highest assigned GT index
            const float* tb = &sgtb[last * 4];
            float q[4 * NBINS];
            const float4* pr4 = (const float4*)(reg + ((size_t)b * A + a) * (4 * NBINS));
#pragma unroll
            for (int i = 0; i < NBINS; ++i) {      // 4*17 = 68 floats = 17 x b128
                float4 t4 = pr4[i];
                q[4 * i + 0] = t4.x; q[4 * i + 1] = t4.y;
                q[4 * i + 2] = t4.z; q[4 * i + 3] = t4.w;
            }
            float ax = anchor_points[2 * a], ay = anchor_points[2 * a + 1];
            float st = stride_t[a];
            float dist[4];
#pragma unroll
            for (int sd = 0; sd < 4; ++sd) {
                const float* qq = q + sd * NBINS;
                float mx = qq[0];
#pragma unroll
                for (int j = 1; j < NBINS; ++j) mx = fmaxf(mx, qq[j]);
                float den = 0.f, num = 0.f;
#pragma unroll
                for (int j = 0; j < NBINS; ++j) {
                    float e = expf(qq[j] - mx);
                    den += e; num += e * (float)j;
                }
                dist[sd] = num / den;
            }
            float pb[4] = { ax - dist[0] * st, ay - dist[1] * st,
                            ax + dist[2] * st, ay + dist[3] * st };
#pragma unroll
            for (int i = 0; i < 4; ++i) {
                float d = pb[i] - tb[i];
                float ad = fabsf(d);
                box += (ad < 1.0f) ? 0.5f * d * d : ad - 0.5f;
            }
        }
    }

    sp[tid] = pos; sn[tid] = neg; sb[tid] = box; sc[tid] = c;
    __syncthreads();
    for (int s = 128; s > 0; s >>= 1) {
        if (tid < s) {
            sp[tid] += sp[tid + s]; sn[tid] += sn[tid + s];
            sb[tid] += sb[tid + s]; sc[tid] += sc[tid + s];
        }
        __syncthreads();
    }
    if (tid == 0) {
        atomicAdd(&accum[b * 3 + 0], sp[0]);
        atomicAdd(&accum[b * 3 + 1], sn[0]);
        atomicAdd(&accum[b * 3 + 2], sb[0]);
        atomicAdd(&cnt[b], sc[0]);
    }
}

// ---------------------------------------------------------------- final
__global__ void final_kernel(const float* accum, const unsigned int* cnt,
                             float* out, int B, int A)
{
    __shared__ float scl[64], sbx[64];
    const int t = threadIdx.x;
    float cl = 0.f, bx = 0.f;
    if (t < B) {
        float np  = (float)cnt[t];
        float pos = accum[t * 3 + 0];
        float neg = accum[t * 3 + 1];
        float bxs = accum[t * 3 + 2];
        float pos_w = fminf((float)A / fmaxf(np, 1.0f), 50.0f);
        float pos_loss = (np > 0.f) ? pos / fmaxf(np * (float)NCLS, 1.0f) * pos_w : 0.f;
        float neg_loss = neg / fmaxf(((float)A - np) * (float)NCLS, 1.0f);
        cl = pos_loss + neg_loss;
        bx = (np > 0.f) ? bxs / fmaxf(np * 4.0f, 1.0f) : 0.f;
    }
    scl[t] = cl; sbx[t] = bx;
    __syncthreads();
    for (int s = 32; s > 0; s >>= 1) {
        if (t < s) { scl[t] += scl[t + s]; sbx[t] += sbx[t + s]; }
        __syncthreads();
    }
    if (t == 0) out[0] = (scl[0] / (float)B) * 1.0f + (sbx[0] / (float)B) * 2.5f;
}

// ---------------------------------------------------------------- launch
extern "C" void kernel_launch(void* const* d_in, const int* in_sizes, int n_in,
                              void* d_out, int out_size, void* d_ws, size_t ws_size,
                              hipStream_t stream) {
    const float* cls = (const float*)d_in[0];
    const float* reg = (const float*)d_in[1];
    const float* gtb = (const float*)d_in[2];
    const float* ap  = (const float*)d_in[3];
    const float* st  = (const float*)d_in[4];
    const int*   gtl = (const int*)d_in[5];

    const int A  = in_sizes[3] / 2;                 // anchor_points (A,2)
    const int BG = in_sizes[5];                     // gt_labels (B,G)
    const int B  = in_sizes[0] / (A * NCLS);        // cls_scores (B,A,C)
    const int G  = BG / B;

    int K = A / (G > 0 ? G : 1);                    // k = clamp(A//G, 3, 13)
    if (K > KMAX) K = KMAX;
    if (K < 3)    K = 3;

    // ws: [mask u32 B*A][accum f32 B*3][cnt u32 B]
    unsigned int* mask  = (unsigned int*)d_ws;
    float*        accum = (float*)((char*)d_ws + (size_t)B * A * 4);
    unsigned int* cnt   = (unsigned int*)((char*)accum + (size_t)B * 3 * 4);

    const int nz = B * A + 4 * B;
    zero_ws_kernel<<<(nz + 255) / 256, 256, 0, stream>>>((unsigned int*)d_ws, nz);

    assign_kernel<<<B * G, 256, 0, stream>>>(gtb, ap, mask, B, G, A, K);

    dim3 g2((A + 255) / 256, B);
    loss_kernel<<<g2, 256, 0, stream>>>(cls, reg, gtb, gtl, ap, st, mask,
                                        accum, cnt, B, G, A);

    final_kernel<<<1, 64, 0, stream>>>(accum, cnt, (float*)d_out, B, A);
}